// TripletCenterLoss_72413148610843
// MI455X (gfx1250) — compile-verified
//
#include <hip/hip_runtime.h>
#include <math.h>

typedef __attribute__((ext_vector_type(2))) float v2f;
typedef __attribute__((ext_vector_type(8))) float v8f;

#define B_N    4096
#define D_K    512
#define C_N    1000
#define MARGIN 0.3f
#define EPS_D2 1e-12f

// ---------------------------------------------------------------------------
// Kernel 1: squared L2 norm of every center row (1000 x 512).
// One wave per center row; lane-strided partial sums + shfl-xor reduction.
// ---------------------------------------------------------------------------
__global__ __launch_bounds__(256) void center_norms_kernel(
    const float* __restrict__ centers, float* __restrict__ c2c) {
  int gwave = (blockIdx.x * blockDim.x + threadIdx.x) >> 5;
  int lane  = threadIdx.x & 31;
  if (gwave >= C_N) return;
  const float* row = centers + (size_t)gwave * D_K;
  float s = 0.0f;
  for (int k = lane; k < D_K; k += 32) {
    float v = row[k];
    s += v * v;
  }
  #pragma unroll
  for (int off = 16; off > 0; off >>= 1) s += __shfl_xor(s, off, 32);
  if (lane == 0) c2c[gwave] = s;
}

// ---------------------------------------------------------------------------
// Kernel 2: fused tiled GEMM (WMMA f32 16x16x4) + distance + masked row
// max/min + margin loss partial sum per 16-row strip.
//   Block  = 256 threads (8 waves), owns 16 rows of `inputs`.
//   Wave w = column tiles w, w+8, ..., 248+w (16 columns each).
// ---------------------------------------------------------------------------
__global__ __launch_bounds__(256) void triplet_tile_kernel(
    const float* __restrict__ inputs,  const int* __restrict__ targets,
    const float* __restrict__ centers, const float* __restrict__ c2c,
    float* __restrict__ blockSums) {
  __shared__ float ldsA[16 * D_K];   // 32 KB input strip (row-major)
  __shared__ float psum[256];
  __shared__ float x2s[16];
  __shared__ int   trow_s[16];
  __shared__ float apP[8][16];
  __shared__ float anP[8][16];

  const int tid = threadIdx.x;
  const int m0  = blockIdx.x * 16;   // first row of this strip

  // Stage the 16x512 input strip into LDS (contiguous in memory).
  {
    const float4* src = (const float4*)(inputs + (size_t)m0 * D_K);
    float4* dst = (float4*)ldsA;
    #pragma unroll
    for (int i = tid; i < (16 * D_K) / 4; i += 256) dst[i] = src[i];
  }
  if (tid < 16) trow_s[tid] = targets[m0 + tid];
  __syncthreads();

  // x2[m] = ||inputs[m]||^2 for the 16 strip rows (deterministic tree).
  {
    int row = tid >> 4, chunk = tid & 15;
    const float* p = &ldsA[row * D_K + chunk * 32];
    float s = 0.0f;
    #pragma unroll
    for (int e = 0; e < 32; ++e) { float v = p[e]; s += v * v; }
    psum[tid] = s;
  }
  __syncthreads();
  if (tid < 16) {
    float s = 0.0f;
    for (int c = 0; c < 16; ++c) s += psum[tid * 16 + c];
    x2s[tid] = s;
  }
  __syncthreads();

  const int wave = tid >> 5;
  const int lane = tid & 31;
  const int hi   = lane >> 4;   // 0: K-pair {0,1} / rows r ; 1: K-pair {2,3} / rows r+8
  const int lo   = lane & 15;   // column-within-tile (B/C/D layout) & row (A layout)

  // Per-lane row-constant data in the C/D layout (rows m = r + 8*hi).
  float x2v[8]; int trv[8];
  #pragma unroll
  for (int r = 0; r < 8; ++r) {
    int m = r + 8 * hi;
    x2v[r] = x2s[m];
    trv[r] = trow_s[m];
  }

  float ap[8], an[8];
  #pragma unroll
  for (int r = 0; r < 8; ++r) { ap[r] = -INFINITY; an[r] = INFINITY; }

  // A-matrix lane view: row = lo, K offset = 2*hi (float2 per WMMA step).
  const float* aLds = &ldsA[lo * D_K + 2 * hi];

  for (int ct = wave; ct < B_N / 16; ct += 8) {
    const int col = ct * 16 + lo;           // output column j
    const int tj  = targets[col];           // gather index for this column
    const float c2j = c2c[tj];
    const float* bp = centers + (size_t)tj * D_K + 2 * hi;

    v8f acc0 = {0}, acc1 = {0}, acc2 = {0}, acc3 = {0};
    #pragma unroll 4
    for (int k0 = 0; k0 < D_K; k0 += 16) {
      v2f a0 = *(const v2f*)(aLds + k0);
      v2f b0 = *(const v2f*)(bp   + k0);
      v2f a1 = *(const v2f*)(aLds + k0 + 4);
      v2f b1 = *(const v2f*)(bp   + k0 + 4);
      v2f a2 = *(const v2f*)(aLds + k0 + 8);
      v2f b2 = *(const v2f*)(bp   + k0 + 8);
      v2f a3 = *(const v2f*)(aLds + k0 + 12);
      v2f b3 = *(const v2f*)(bp   + k0 + 12);
      acc0 = __builtin_amdgcn_wmma_f32_16x16x4_f32(false, a0, false, b0,
                                                   (short)0, acc0, false, false);
      acc1 = __builtin_amdgcn_wmma_f32_16x16x4_f32(false, a1, false, b1,
                                                   (short)0, acc1, false, false);
      acc2 = __builtin_amdgcn_wmma_f32_16x16x4_f32(false, a2, false, b2,
                                                   (short)0, acc2, false, false);
      acc3 = __builtin_amdgcn_wmma_f32_16x16x4_f32(false, a3, false, b3,
                                                   (short)0, acc3, false, false);
    }
    v8f acc = acc0 + acc1 + acc2 + acc3;

    // Epilogue in C/D layout: lane holds column `col`, rows r+8*hi.
    #pragma unroll
    for (int r = 0; r < 8; ++r) {
      float d2   = x2v[r] + c2j - 2.0f * acc[r];
      float dist = sqrtf(fmaxf(d2, EPS_D2));
      if (trv[r] == tj) ap[r] = fmaxf(ap[r], dist);
      else              an[r] = fminf(an[r], dist);
    }
  }

  // Reduce across the 16 lanes holding the same rows (columns of the tile).
  #pragma unroll
  for (int r = 0; r < 8; ++r) {
    #pragma unroll
    for (int off = 1; off < 16; off <<= 1) {
      ap[r] = fmaxf(ap[r], __shfl_xor(ap[r], off, 32));
      an[r] = fminf(an[r], __shfl_xor(an[r], off, 32));
    }
  }
  if (lo == 0) {
    #pragma unroll
    for (int r = 0; r < 8; ++r) {
      apP[wave][r + 8 * hi] = ap[r];
      anP[wave][r + 8 * hi] = an[r];
    }
  }
  __syncthreads();

  // Combine the 8 waves' column-partials, apply margin + relu per row.
  if (tid < 16) {
    float a = -INFINITY, n = INFINITY;
    for (int w = 0; w < 8; ++w) {
      a = fmaxf(a, apP[w][tid]);
      n = fminf(n, anP[w][tid]);
    }
    float l = a - n + MARGIN;        // an may be +inf -> l = -inf -> relu 0
    psum[tid] = (l > 0.0f) ? l : 0.0f;
  }
  __syncthreads();
  if (tid == 0) {
    float s = 0.0f;
    for (int i = 0; i < 16; ++i) s += psum[i];
    blockSums[blockIdx.x] = s;
  }
}

// ---------------------------------------------------------------------------
// Kernel 3: deterministic reduction of the 256 strip sums -> mean loss.
// ---------------------------------------------------------------------------
__global__ __launch_bounds__(256) void finalize_kernel(
    const float* __restrict__ blockSums, float* __restrict__ out) {
  __shared__ float s[256];
  int t = threadIdx.x;
  s[t] = blockSums[t];
  __syncthreads();
  for (int off = 128; off > 0; off >>= 1) {
    if (t < off) s[t] += s[t + off];
    __syncthreads();
  }
  if (t == 0) out[0] = s[0] / (float)B_N;
}

extern "C" void kernel_launch(void* const* d_in, const int* in_sizes, int n_in,
                              void* d_out, int out_size, void* d_ws, size_t ws_size,
                              hipStream_t stream) {
  const float* inputs  = (const float*)d_in[0];
  const int*   targets = (const int*)  d_in[1];
  const float* centers = (const float*)d_in[2];

  float* c2c       = (float*)d_ws;        // 1000 floats
  float* blockSums = c2c + C_N;           // 256 floats

  center_norms_kernel<<<(C_N + 7) / 8, 256, 0, stream>>>(centers, c2c);
  triplet_tile_kernel<<<B_N / 16, 256, 0, stream>>>(inputs, targets, centers,
                                                    c2c, blockSums);
  finalize_kernel<<<1, 256, 0, stream>>>(blockSums, (float*)d_out);
}